// EdgeGAT_4896262718106
// MI455X (gfx1250) — compile-verified
//
#include <hip/hip_runtime.h>
#include <hip/hip_bf16.h>
#include <math.h>

// ---------------------------------------------------------------------------
// EdgeGAT on MI455X (gfx1250): fp32 WMMA (V_WMMA_F32_16X16X4_F32) for all
// GEMMs, algebraic fusion of the edge-MLP + fc_edge branch, scalarized node
// logits, atomic-based edge softmax/aggregation.
// ---------------------------------------------------------------------------

typedef __attribute__((ext_vector_type(2))) float v2f;
typedef __attribute__((ext_vector_type(8))) float v8f;

#define HGAT 8
#define CGAT 64
#define DGAT 512

static inline int cdiv(long long a, long long b) { return (int)((a + b - 1) / b); }

// ---------------- fills ----------------
__global__ void fill_f32(float* p, float v, size_t n) {
  size_t i = blockIdx.x * (size_t)blockDim.x + threadIdx.x;
  if (i < n) p[i] = v;
}
__global__ void fill_u32(unsigned* p, unsigned v, size_t n) {
  size_t i = blockIdx.x * (size_t)blockDim.x + threadIdx.x;
  if (i < n) p[i] = v;
}

// ---------------- generic GEMM: C[M,512] = A[M,K] x W[K,512] ----------------
// Block = 4 waves (32x4). blockIdx.y selects the 16-row M tile (shared A tile
// staged in LDS, padded to kill bank conflicts); each wave owns one 16-col
// N tile. Inner loop: V_WMMA_F32_16X16X4_F32 over K in steps of 4.
template <int K>
__global__ void gemm_wmma(const float* __restrict__ A, const float* __restrict__ W,
                          float* __restrict__ Cout, int M) {
  __shared__ float sA[16 * (K + 1)];
  const int lane = threadIdx.x;
  const int wv = threadIdx.y;           // 0..3
  const int tid = wv * 32 + lane;
  const int m0 = blockIdx.y * 16;
  const int n0 = (blockIdx.x * 4 + wv) * 16;

  for (int idx = tid; idx < 16 * K; idx += 128) {
    int r = idx / K, c = idx - r * K;
    int row = m0 + r; if (row >= M) row = M - 1;   // clamp (stores predicated)
    sA[r * (K + 1) + c] = A[(size_t)row * K + c];
  }
  __syncthreads();

  const int half = lane >> 4;   // A: lanes 0-15 hold K+0/K+1, lanes 16-31 K+2/K+3
  const int l = lane & 15;

  v8f acc = {};
  for (int k0 = 0; k0 < K; k0 += 4) {
    v2f a, b;
    a.x = sA[l * (K + 1) + k0 + 2 * half];
    a.y = sA[l * (K + 1) + k0 + 2 * half + 1];
    b.x = W[(size_t)(k0 + 2 * half) * DGAT + n0 + l];
    b.y = W[(size_t)(k0 + 2 * half + 1) * DGAT + n0 + l];
    acc = __builtin_amdgcn_wmma_f32_16x16x4_f32(false, a, false, b, (short)0, acc,
                                                false, false);
  }
#pragma unroll
  for (int r = 0; r < 8; ++r) {              // C/D: VGPR r -> M=r (lo) / r+8 (hi)
    int m = m0 + r + 8 * half;
    if (m < M) Cout[(size_t)m * DGAT + n0 + l] = acc[r];
  }
}

// ---------------- attention scores: a_s/a_d [N,H] ----------------
__global__ void att_scores(const float* __restrict__ h, const float* __restrict__ as_w,
                           const float* __restrict__ ad_w, float* __restrict__ a_s,
                           float* __restrict__ a_d, int N) {
  int idx = blockIdx.x * blockDim.x + threadIdx.x;
  if (idx >= N * HGAT) return;
  int n = idx / HGAT, hd = idx - n * HGAT;
  const float* hp = h + (size_t)n * DGAT + hd * CGAT;
  const float* ws = as_w + hd * CGAT;
  const float* wd = ad_w + hd * CGAT;
  float s = 0.f, d = 0.f;
#pragma unroll 8
  for (int c = 0; c < CGAT; ++c) { float v = hp[c]; s += v * ws[c]; d += v * wd[c]; }
  a_s[idx] = s; a_d[idx] = d;
}

// ---------------- edge softmax machinery ----------------
__device__ __forceinline__ void edge_ends(const int* __restrict__ ei, int E, int et,
                                          int& s, int& d) {
  if (et < E) { s = ei[et]; d = ei[E + et]; } else { s = d = et - E; }  // self loops
}
__device__ __forceinline__ float lrelu02(float x) { return x > 0.f ? x : 0.2f * x; }
__device__ __forceinline__ unsigned fkey(float x) {          // monotone float->uint
  unsigned b = __float_as_uint(x);
  return (b & 0x80000000u) ? ~b : (b | 0x80000000u);
}
__device__ __forceinline__ float funkey(unsigned k) {
  return __uint_as_float((k & 0x80000000u) ? (k & 0x7fffffffu) : ~k);
}

__global__ void edge_max(const int* __restrict__ ei, const float* __restrict__ a_s,
                         const float* __restrict__ a_d, unsigned* __restrict__ mk,
                         int E, int N) {
  int idx = blockIdx.x * blockDim.x + threadIdx.x;
  int ET = E + N;
  if (idx >= ET * HGAT) return;
  int et = idx >> 3, hd = idx & 7;
  int s, d; edge_ends(ei, E, et, s, d);
  float e = lrelu02(a_s[s * HGAT + hd] + a_d[d * HGAT + hd]);
  atomicMax(&mk[d * HGAT + hd], fkey(e));
}

__global__ void edge_expsum(const int* __restrict__ ei, const float* __restrict__ a_s,
                            const float* __restrict__ a_d, const unsigned* __restrict__ mk,
                            float* __restrict__ ex, float* __restrict__ ssum,
                            int E, int N) {
  int idx = blockIdx.x * blockDim.x + threadIdx.x;
  int ET = E + N;
  if (idx >= ET * HGAT) return;
  int et = idx >> 3, hd = idx & 7;
  int s, d; edge_ends(ei, E, et, s, d);
  float e = lrelu02(a_s[s * HGAT + hd] + a_d[d * HGAT + hd]);
  float m = funkey(mk[d * HGAT + hd]);
  float v = __expf(e - m);
  ex[idx] = v;
  atomicAdd(&ssum[d * HGAT + hd], v);
}

// out[dst] += h[src] * alpha   (one wave per edge; coalesced 512-float rows)
__global__ void aggregate(const int* __restrict__ ei, const float* __restrict__ h,
                          const float* __restrict__ ex, const float* __restrict__ ssum,
                          float* __restrict__ out, int E, int N) {
  int et = blockIdx.x * blockDim.y + threadIdx.y;
  int ET = E + N;
  if (et >= ET) return;
  int s, d; edge_ends(ei, E, et, s, d);
  int lane = threadIdx.x;
  const float* hs = h + (size_t)s * DGAT;
  float* od = out + (size_t)d * DGAT;
  const float* exe = ex + (size_t)et * HGAT;
  const float* sd = ssum + (size_t)d * HGAT;
  for (int i = lane; i < DGAT; i += 32) {
    int hd = i >> 6;
    float alpha = exe[hd] / (sd[hd] + 1e-16f);
    atomicAdd(&od[i], hs[i] * alpha);
  }
}

// ---------------- bias + ELU, batch-norm ----------------
__global__ void bias_elu(float* __restrict__ Z, const float* __restrict__ bias, int M) {
  size_t i = blockIdx.x * (size_t)blockDim.x + threadIdx.x;
  if (i >= (size_t)M * DGAT) return;
  float x = Z[i] + bias[i & (DGAT - 1)];
  Z[i] = x > 0.f ? x : (__expf(x) - 1.f);
}

__global__ void col_stats(const float* __restrict__ Z, float* __restrict__ s1,
                          float* __restrict__ s2, int M) {
  int col = threadIdx.x;             // 256 threads -> cols col, col+256
  int r0 = blockIdx.x * 64;
  float a0 = 0.f, b0 = 0.f, a1 = 0.f, b1 = 0.f;
  for (int r = 0; r < 64; ++r) {
    int row = r0 + r; if (row >= M) break;
    float x0 = Z[(size_t)row * DGAT + col];
    float x1 = Z[(size_t)row * DGAT + col + 256];
    a0 += x0; b0 += x0 * x0; a1 += x1; b1 += x1 * x1;
  }
  atomicAdd(&s1[col], a0);       atomicAdd(&s2[col], b0);
  atomicAdd(&s1[col + 256], a1); atomicAdd(&s2[col + 256], b1);
}

__global__ void bn_apply(float* __restrict__ Z, const float* __restrict__ s1,
                         const float* __restrict__ s2, const float* __restrict__ g,
                         const float* __restrict__ b, int M) {
  size_t i = blockIdx.x * (size_t)blockDim.x + threadIdx.x;
  if (i >= (size_t)M * DGAT) return;
  int col = i & (DGAT - 1);
  float inv = 1.f / (float)M;
  float mu = s1[col] * inv;
  float var = s2[col] * inv - mu * mu;
  Z[i] = g[col] * (Z[i] - mu) * rsqrtf(var + 1e-5f) + b[col];
}

// ---------------- fused edge-MLP branch ----------------
// v = mlp_w2 @ fc_w_e ;  cedge = mlp_b2 . fc_w_e
__global__ void compute_v(const float* __restrict__ w2, const float* __restrict__ fce,
                          float* __restrict__ v) {
  int i = blockIdx.x * blockDim.x + threadIdx.x;
  if (i >= DGAT) return;
  const float* row = w2 + (size_t)i * DGAT;
  float acc = 0.f;
#pragma unroll 8
  for (int j = 0; j < DGAT; ++j) acc += row[j] * fce[j];
  v[i] = acc;
}
__global__ void dot512(const float* __restrict__ a, const float* __restrict__ b,
                       float* __restrict__ out) {
  __shared__ float red[256];
  int t = threadIdx.x;
  red[t] = a[t] * b[t] + a[t + 256] * b[t + 256];
  __syncthreads();
  for (int s = 128; s; s >>= 1) { if (t < s) red[t] += red[t + s]; __syncthreads(); }
  if (t == 0) out[0] = red[0];
}

// contrib[e] = relu(edge_attr[e] @ W1 + b1) . v + cedge   (WMMA-fused, K=64)
__global__ void edge_mlp_wmma(const float* __restrict__ EA, const float* __restrict__ W1,
                              const float* __restrict__ B1, const float* __restrict__ V,
                              const float* __restrict__ cedge, float* __restrict__ out,
                              int E) {
  const int K = 64;
  __shared__ float sA[16 * (K + 1)];
  __shared__ float red[16];
  const int lane = threadIdx.x, wv = threadIdx.y;   // 8 waves
  const int tid = wv * 32 + lane;
  const int e0 = blockIdx.x * 16;

  for (int idx = tid; idx < 16 * K; idx += 256) {
    int r = idx >> 6, c = idx & 63;
    int er = e0 + r; if (er >= E) er = E - 1;
    sA[r * (K + 1) + c] = EA[(size_t)er * K + c];
  }
  if (tid < 16) red[tid] = 0.f;
  __syncthreads();

  const int half = lane >> 4, l = lane & 15;
  for (int t = 0; t < 4; ++t) {                 // wave covers 4 of 32 N-tiles
    int n0 = (wv + t * 8) * 16;
    v8f acc = {};
    for (int k0 = 0; k0 < K; k0 += 4) {
      v2f a, b;
      a.x = sA[l * (K + 1) + k0 + 2 * half];
      a.y = sA[l * (K + 1) + k0 + 2 * half + 1];
      b.x = W1[(size_t)(k0 + 2 * half) * DGAT + n0 + l];
      b.y = W1[(size_t)(k0 + 2 * half + 1) * DGAT + n0 + l];
      acc = __builtin_amdgcn_wmma_f32_16x16x4_f32(false, a, false, b, (short)0, acc,
                                                  false, false);
    }
    int n = n0 + l;
    float b1 = B1[n], vv = V[n];
#pragma unroll
    for (int r = 0; r < 8; ++r) {
      float x = acc[r] + b1;
      float part = (x > 0.f ? x : 0.f) * vv;
      part += __shfl_xor(part, 1);
      part += __shfl_xor(part, 2);
      part += __shfl_xor(part, 4);
      part += __shfl_xor(part, 8);              // 16-lane half-wave sum
      if (l == 0) atomicAdd(&red[r + 8 * half], part);
    }
  }
  __syncthreads();
  if (tid < 16 && e0 + tid < E) out[e0 + tid] = red[tid] + cedge[0];
}

// ---------------- node logits + final gather ----------------
__global__ void node_logits(const float* __restrict__ h, const float* __restrict__ wsrc,
                            const float* __restrict__ wdst, float* __restrict__ nls,
                            float* __restrict__ nld, int N) {
  int lane = threadIdx.x;
  int n = blockIdx.x * blockDim.y + threadIdx.y;
  if (n >= N) return;
  const float* hp = h + (size_t)n * DGAT;
  float s = 0.f, d = 0.f;
  for (int i = lane; i < DGAT; i += 32) { float v = hp[i]; s += v * wsrc[i]; d += v * wdst[i]; }
  for (int m = 16; m; m >>= 1) { s += __shfl_xor(s, m); d += __shfl_xor(d, m); }
  if (lane == 0) { nls[n] = s; nld[n] = d; }
}

__global__ void final_logits(const int* __restrict__ ei, const float* __restrict__ nls,
                             const float* __restrict__ nld, const float* __restrict__ contrib,
                             const float* __restrict__ fcb, float* __restrict__ out, int E) {
  int e = blockIdx.x * blockDim.x + threadIdx.x;
  if (e >= E) return;
  out[e] = nls[ei[e]] + nld[ei[E + e]] + contrib[e] + fcb[0];
}

// ---------------------------------------------------------------------------
extern "C" void kernel_launch(void* const* d_in, const int* in_sizes, int n_in,
                              void* d_out, int out_size, void* d_ws, size_t ws_size,
                              hipStream_t stream) {
  // setup_inputs order: x, edge_index, edge_attr, then params (pytree / sorted
  // dict keys): bn1_b, bn1_g, bn2_b, bn2_g, bn3_b, bn3_g, conv{1,2,3}{W,
  // att_dst, att_src, bias}, fc_b, fc_w_dst, fc_w_e, fc_w_src, mlp_b1, mlp_b2,
  // mlp_w1, mlp_w2.
  const float* x  = (const float*)d_in[0];
  const int*   ei = (const int*)d_in[1];
  const float* ea = (const float*)d_in[2];
  const float* bn_b[3] = { (const float*)d_in[3], (const float*)d_in[5], (const float*)d_in[7] };
  const float* bn_g[3] = { (const float*)d_in[4], (const float*)d_in[6], (const float*)d_in[8] };
  const float* cW[3]  = { (const float*)d_in[9],  (const float*)d_in[13], (const float*)d_in[17] };
  const float* cAd[3] = { (const float*)d_in[10], (const float*)d_in[14], (const float*)d_in[18] };
  const float* cAs[3] = { (const float*)d_in[11], (const float*)d_in[15], (const float*)d_in[19] };
  const float* cB[3]  = { (const float*)d_in[12], (const float*)d_in[16], (const float*)d_in[20] };
  const float* fc_b     = (const float*)d_in[21];
  const float* fc_w_dst = (const float*)d_in[22];
  const float* fc_w_e   = (const float*)d_in[23];
  const float* fc_w_src = (const float*)d_in[24];
  const float* mlp_b1   = (const float*)d_in[25];
  const float* mlp_b2   = (const float*)d_in[26];
  const float* mlp_w1   = (const float*)d_in[27];
  const float* mlp_w2   = (const float*)d_in[28];

  const int FN = 128;
  const int N  = in_sizes[0] / FN;      // 25000
  const int E  = in_sizes[1] / 2;       // 250000
  const int ET = E + N;                 // with self loops

  // workspace layout (floats)
  float* ws = (float*)d_ws;
  float* h0 = ws;                                   // N*512
  float* h1 = h0 + (size_t)N * DGAT;                // N*512
  float* a_s = h1 + (size_t)N * DGAT;               // N*8
  float* a_d = a_s + (size_t)N * HGAT;              // N*8
  unsigned* mk = (unsigned*)(a_d + (size_t)N * HGAT); // N*8
  float* ssum = (float*)(mk + (size_t)N * HGAT);    // N*8
  float* ex   = ssum + (size_t)N * HGAT;            // ET*8
  float* csum = ex + (size_t)ET * HGAT;             // 512
  float* csq  = csum + DGAT;                        // 512
  float* vvec = csq + DGAT;                         // 512
  float* cedge = vvec + DGAT;                       // 1
  float* nls = cedge + 1;                           // N
  float* nld = nls + N;                             // N
  float* contrib = nld + N;                         // E

  dim3 blk32x4(32, 4), blk32x8(32, 8);
  const size_t nD = (size_t)N * DGAT;

  const float* in = x;
  float* hcur = h0;
  float* hagg = h1;

  for (int L = 0; L < 3; ++L) {
    // h = in @ W   (fp32 WMMA)
    if (L == 0)
      gemm_wmma<128><<<dim3(DGAT / 64, cdiv(N, 16)), blk32x4, 0, stream>>>(in, cW[L], hcur, N);
    else
      gemm_wmma<512><<<dim3(DGAT / 64, cdiv(N, 16)), blk32x4, 0, stream>>>(in, cW[L], hcur, N);

    att_scores<<<cdiv((long long)N * HGAT, 256), 256, 0, stream>>>(hcur, cAs[L], cAd[L], a_s, a_d, N);

    fill_u32<<<cdiv((long long)N * HGAT, 256), 256, 0, stream>>>(mk, 0u, (size_t)N * HGAT);
    fill_f32<<<cdiv((long long)N * HGAT, 256), 256, 0, stream>>>(ssum, 0.f, (size_t)N * HGAT);
    fill_f32<<<cdiv(nD, 256), 256, 0, stream>>>(hagg, 0.f, nD);

    edge_max<<<cdiv((long long)ET * HGAT, 256), 256, 0, stream>>>(ei, a_s, a_d, mk, E, N);
    edge_expsum<<<cdiv((long long)ET * HGAT, 256), 256, 0, stream>>>(ei, a_s, a_d, mk, ex, ssum, E, N);
    aggregate<<<cdiv(ET, 8), blk32x8, 0, stream>>>(ei, hcur, ex, ssum, hagg, E, N);

    bias_elu<<<cdiv(nD, 256), 256, 0, stream>>>(hagg, cB[L], N);
    fill_f32<<<cdiv(DGAT, 256), 256, 0, stream>>>(csum, 0.f, DGAT);
    fill_f32<<<cdiv(DGAT, 256), 256, 0, stream>>>(csq, 0.f, DGAT);
    col_stats<<<cdiv(N, 64), 256, 0, stream>>>(hagg, csum, csq, N);
    bn_apply<<<cdiv(nD, 256), 256, 0, stream>>>(hagg, csum, csq, bn_g[L], bn_b[L], N);

    in = hagg;                      // next layer input
    float* t = hcur; hcur = hagg; hagg = t;
  }
  const float* hfin = in;           // final node features [N,512]

  // fused edge branch: contrib[e] = relu(ea@W1+b1) . (W2@fc_w_e) + b2.fc_w_e
  compute_v<<<2, 256, 0, stream>>>(mlp_w2, fc_w_e, vvec);
  dot512<<<1, 256, 0, stream>>>(mlp_b2, fc_w_e, cedge);
  edge_mlp_wmma<<<cdiv(E, 16), blk32x8, 0, stream>>>(ea, mlp_w1, mlp_b1, vvec, cedge, contrib, E);

  // scalar node logits then per-edge gather
  node_logits<<<cdiv(N, 8), blk32x8, 0, stream>>>(hfin, fc_w_src, fc_w_dst, nls, nld, N);
  final_logits<<<cdiv(E, 256), 256, 0, stream>>>(ei, nls, nld, contrib, fc_b, (float*)d_out, E);
}